// Mainmodel_66546223284649
// MI455X (gfx1250) — compile-verified
//
#include <hip/hip_runtime.h>

typedef float v2f __attribute__((ext_vector_type(2)));
typedef float v8f __attribute__((ext_vector_type(8)));

#define NUM_CTX 20
#define DIM     64
#define ROWS_PER_WAVE 16
#define WAVES_PER_BLOCK 8
#define ROWS_PER_BLOCK (ROWS_PER_WAVE * WAVES_PER_BLOCK)   // 128

// Fused: n2 = ||x||^2, dots = x @ x_ctx^T via V_WMMA_F32_16X16X4_F32,
// d = sqrt(max(n2 + n2c - 2*dot, 0)), partial[b] = sum |d - dis| for this block.
__global__ void __launch_bounds__(256)
pdist_loss_wmma(const float* __restrict__ x,
                const float* __restrict__ dis,
                float* __restrict__ partial,
                int N)
{
    // 32 x 64 padded context tile (rows 20..31 zero) — 8 KB LDS
    __shared__ float ctx[32 * DIM];
    __shared__ float wavesum[WAVES_PER_BLOCK];

    const int tid = threadIdx.x;

    // Cooperative, coalesced fill of ctx (first 20 rows of x, rest zero)
    for (int i = tid; i < 32 * DIM; i += 256) {
        int r = i >> 6;
        ctx[i] = (r < NUM_CTX) ? x[i] : 0.0f;   // ctx rows are x rows 0..19, same layout
    }
    __syncthreads();

    const int wave     = tid >> 5;          // wave32
    const int lane     = tid & 31;
    const int laneHalf = lane >> 4;         // which K-half this lane carries
    const int lmod     = lane & 15;

    const long rowBase    = (long)blockIdx.x * ROWS_PER_BLOCK + (long)wave * ROWS_PER_WAVE;
    const long myRow      = rowBase + lmod; // A-fragment row carried by this lane
    const long rowClamped = (myRow < N) ? myRow : (long)(N - 1);
    const float* __restrict__ arow = x + rowClamped * DIM;

    // Prefetch this wave's dis rows (streamed later in the epilogue)
    if (rowClamped * NUM_CTX < (long)N * NUM_CTX)
        __builtin_prefetch(&dis[rowClamped * NUM_CTX], 0, 0);

    v8f c0 = {};   // dots vs ctx cols  0..15
    v8f c1 = {};   // dots vs ctx cols 16..31 (20..31 are zero pad)
    float n2p = 0.0f, n2c0p = 0.0f, n2c1p = 0.0f;

    // K = 64 in chunks of 4; lane carries K = 4*kk + 2*laneHalf + {0,1}
    #pragma unroll
    for (int kk = 0; kk < DIM / 4; ++kk) {
        const int k0 = kk * 4 + laneHalf * 2;
        const v2f a  = *(const v2f*)(arow + k0);
        const v2f b0 = *(const v2f*)(&ctx[lmod * DIM + k0]);
        const v2f b1 = *(const v2f*)(&ctx[(16 + lmod) * DIM + k0]);

        n2p   = fmaf(a.x,  a.x,  fmaf(a.y,  a.y,  n2p));
        n2c0p = fmaf(b0.x, b0.x, fmaf(b0.y, b0.y, n2c0p));
        n2c1p = fmaf(b1.x, b1.x, fmaf(b1.y, b1.y, n2c1p));

        // D = A(16x4) * B(4x16) + C   — fp32 WMMA, chained accumulator
        c0 = __builtin_amdgcn_wmma_f32_16x16x4_f32(false, a, false, b0,
                                                   (short)0, c0, false, false);
        c1 = __builtin_amdgcn_wmma_f32_16x16x4_f32(false, a, false, b1,
                                                   (short)0, c1, false, false);
    }

    // Each lane held half of the K range: combine with its partner lane (^16)
    const float n2row = n2p   + __shfl_xor(n2p,   16, 32); // norm of row rowBase+lmod (lanes L and L+16)
    const float n2c0  = n2c0p + __shfl_xor(n2c0p, 16, 32); // norm of ctx col lmod
    const float n2c1  = n2c1p + __shfl_xor(n2c1p, 16, 32); // norm of ctx col 16+lmod

    // C layout: lane = column (lmod), VGPR v = row v + 8*laneHalf
    float acc = 0.0f;
    #pragma unroll
    for (int v = 0; v < 8; ++v) {
        const int  m    = v + 8 * laneHalf;        // local row of this C element
        const float n2m = __shfl(n2row, m, 32);    // lane m holds norm of local row m
        const long grow = rowBase + m;
        if (grow < N) {
            const float* __restrict__ drow = dis + grow * NUM_CTX;
            // tile 0: context column n = lmod (always < 16 < NUM_CTX)
            {
                float sq = fmaxf(n2m + n2c0 - 2.0f * c0[v], 0.0f);
                float d  = (sq > 0.0f) ? sqrtf(sq) : 0.0f;
                acc += fabsf(d - drow[lmod]);
            }
            // tile 1: context column n = 16 + lmod (valid only for lmod < 4)
            if (16 + lmod < NUM_CTX) {
                float sq = fmaxf(n2m + n2c1 - 2.0f * c1[v], 0.0f);
                float d  = (sq > 0.0f) ? sqrtf(sq) : 0.0f;
                acc += fabsf(d - drow[16 + lmod]);
            }
        }
    }

    // Deterministic wave-level tree reduction
    #pragma unroll
    for (int off = 16; off >= 1; off >>= 1)
        acc += __shfl_xor(acc, off, 32);
    if (lane == 0) wavesum[wave] = acc;
    __syncthreads();

    if (tid == 0) {
        float s = 0.0f;
        #pragma unroll
        for (int w = 0; w < WAVES_PER_BLOCK; ++w) s += wavesum[w];
        partial[blockIdx.x] = s;
    }
}

// Deterministic single-block final reduction: out = sum(partial) / N
__global__ void __launch_bounds__(256)
reduce_partials(const float* __restrict__ partial, int nblocks,
                float* __restrict__ out, float invN)
{
    __shared__ float sh[256];
    float s = 0.0f;
    for (int i = threadIdx.x; i < nblocks; i += 256)
        s += partial[i];
    sh[threadIdx.x] = s;
    __syncthreads();
    #pragma unroll
    for (int st = 128; st > 0; st >>= 1) {
        if (threadIdx.x < st) sh[threadIdx.x] += sh[threadIdx.x + st];
        __syncthreads();
    }
    if (threadIdx.x == 0) out[0] = sh[0] * invN;
}

extern "C" void kernel_launch(void* const* d_in, const int* in_sizes, int n_in,
                              void* d_out, int out_size, void* d_ws, size_t ws_size,
                              hipStream_t stream)
{
    const float* x   = (const float*)d_in[0];   // (N, 64) f32
    const float* dis = (const float*)d_in[1];   // (N, 20) f32
    float* out       = (float*)d_out;           // scalar loss
    float* partial   = (float*)d_ws;            // per-block partial sums

    const int N        = in_sizes[0] / DIM;     // 200000
    const int nblocks  = (N + ROWS_PER_BLOCK - 1) / ROWS_PER_BLOCK;

    pdist_loss_wmma<<<nblocks, 256, 0, stream>>>(x, dis, partial, N);
    reduce_partials<<<1, 256, 0, stream>>>(partial, nblocks, out, 1.0f / (float)N);
}